// RotationKernel_59889023975888
// MI455X (gfx1250) — compile-verified
//
#include <hip/hip_runtime.h>

typedef __attribute__((ext_vector_type(2))) float v2f;
typedef __attribute__((ext_vector_type(8))) float v8f;

#define DIMS 16

// Pass 1: row squared-norms for X1 and X2 -> workspace
__global__ __launch_bounds__(256) void sqnorms_kernel(const float* __restrict__ X1,
                                                      const float* __restrict__ X2,
                                                      float* __restrict__ sq,
                                                      int N1, int N2) {
    int i = blockIdx.x * blockDim.x + threadIdx.x;
    const float* src;
    int r;
    if (i < N1)            { src = X1; r = i; }
    else if (i < N1 + N2)  { src = X2; r = i - N1; }
    else return;
    const float4* p = (const float4*)(src + (size_t)r * DIMS);
    float s = 0.0f;
#pragma unroll
    for (int j = 0; j < 4; ++j) {
        float4 v = p[j];
        s += v.x * v.x + v.y * v.y + v.z * v.z + v.w * v.w;
    }
    sq[i] = s;
}

// Pass 2: WMMA f32 16x16x4 GEMM tiles + periodic-kernel epilogue.
// Block = 256 threads = 8 waves. Each wave computes a 16(M) x 64(N) strip:
// 4 independent 16x16 accumulators, K=16 via 4 chained x4 WMMAs each.
__global__ __launch_bounds__(256) void periodic_wmma_kernel(
        const float* __restrict__ X1, const float* __restrict__ X2,
        const float* __restrict__ sq1, const float* __restrict__ sq2,
        const float* __restrict__ beta_p, const float* __restrict__ ls_p,
        float* __restrict__ out, int N1, int N2) {
    const int lane = threadIdx.x & 31;
    const int wave = threadIdx.x >> 5;
    const int l16  = lane & 15;
    const int half = lane >> 4;     // 0: K={0,1} pair, 1: K={2,3} pair (per WMMA A/B layout)
    const int kb   = half * 2;

    const int row0 = (blockIdx.y * 8 + wave) * 16;
    const int col0 = blockIdx.x * 64;
    if (row0 >= N1 || col0 >= N2) return;   // wave-uniform; EXEC stays all-1s for WMMA

    // A chunks: lane covers row row0+l16; chunk c holds k = 4c + kb + {0,1}
    v2f a[4];
    {
        const float* ap = X1 + (size_t)(row0 + l16) * DIMS + kb;
#pragma unroll
        for (int c = 0; c < 4; ++c)
            a[c] = *(const v2f*)(ap + 4 * c);
    }
    // B chunks: B[k][n] = X2[n][k]; same per-lane pattern, 4 tiles along N
    v2f b[4][4];
#pragma unroll
    for (int t = 0; t < 4; ++t) {
        const float* bp = X2 + (size_t)(col0 + 16 * t + l16) * DIMS + kb;
#pragma unroll
        for (int c = 0; c < 4; ++c)
            b[t][c] = *(const v2f*)(bp + 4 * c);
    }

    v8f acc[4] = {};   // zero-init accumulators
    // c outer / t inner: 4 independent WMMAs between dependent accumulations (ILP)
#pragma unroll
    for (int c = 0; c < 4; ++c) {
#pragma unroll
        for (int t = 0; t < 4; ++t) {
            acc[t] = __builtin_amdgcn_wmma_f32_16x16x4_f32(
                /*neg_a=*/false, a[c], /*neg_b=*/false, b[t][c],
                /*c_mod=*/(short)0, acc[t], /*reuse_a=*/false, /*reuse_b=*/false);
        }
    }

    // C/D layout: VGPR v -> M = row0 + v + 8*half, lane%16 -> N
    float sm[8];
#pragma unroll
    for (int v = 0; v < 8; ++v)
        sm[v] = sq1[row0 + v + 8 * half];

    const float beta = beta_p[0];
    const float ls   = ls_p[0];
    const float c2   = -2.0f / (ls * ls);

#pragma unroll
    for (int t = 0; t < 4; ++t) {
        const int n   = col0 + 16 * t + l16;
        const float sn = sq2[n];
        float* op = out + n;
#pragma unroll
        for (int v = 0; v < 8; ++v) {
            float d2   = fmaxf(sm[v] + sn - 2.0f * acc[t][v], 0.0f);
            float dist = sqrtf(d2);
            float s    = __sinf(dist);
            float kv   = beta * __expf(c2 * s * s);
            op[(size_t)(row0 + v + 8 * half) * N2] = kv;
        }
    }
}

extern "C" void kernel_launch(void* const* d_in, const int* in_sizes, int n_in,
                              void* d_out, int out_size, void* d_ws, size_t ws_size,
                              hipStream_t stream) {
    const float* X1     = (const float*)d_in[0];
    const float* X2     = (const float*)d_in[1];
    const float* beta   = (const float*)d_in[2];
    const float* ls     = (const float*)d_in[3];
    float*       out    = (float*)d_out;

    const int N1 = in_sizes[0] / DIMS;   // 8192
    const int N2 = in_sizes[1] / DIMS;   // 8192

    float* sq  = (float*)d_ws;           // [N1] X1 norms, then [N2] X2 norms
    const int tot = N1 + N2;
    sqnorms_kernel<<<(tot + 255) / 256, 256, 0, stream>>>(X1, X2, sq, N1, N2);

    dim3 grid(N2 / 64, N1 / 128);        // 128 x 64 blocks; 8 waves/block, 16x64 strip/wave
    periodic_wmma_kernel<<<grid, dim3(256), 0, stream>>>(
        X1, X2, sq, sq + N1, beta, ls, out, N1, N2);
}